// CustomMultiHeadAttention_63960652972156
// MI455X (gfx1250) — compile-verified
//
#include <hip/hip_runtime.h>
#include <math.h>

// ---------------- problem constants ----------------
#define MDIM 512
#define DDIM 256
#define NB_X 64
#define NB_Y 8
#define NB_ALL 72
#define MD (MDIM * DDIM)           // 131072 floats per batch matrix
#define QKV_SZ ((long)NB_ALL * MD) // 9,437,184 floats per Q/K/V
#define NEG_FILLF (-10000000000.0f)

// output region offsets (floats), in reference return order
#define OFF_CTX_X 0L
#define OFF_CTX_Y 8388608L
#define OFF_SX    9437184L
#define OFF_SY    26214400L
#define OFF_PXY   28311552L
#define OFF_MXY   45088768L
#define OFF_YLEN  61865984L

typedef __attribute__((ext_vector_type(2))) float v2f;
typedef __attribute__((ext_vector_type(8))) float v8f;

static __device__ __forceinline__ v8f wmma_f32(v2f a, v2f b, v8f c) {
    // full-precision f32 WMMA: D(16x16) = A(16x4) * B(4x16) + C
    return __builtin_amdgcn_wmma_f32_16x16x4_f32(false, a, false, b, (short)0, c,
                                                 false, false);
}

// ---------------------------------------------------------------------------
// 1) QKV projection: out[b,i,n] = sum_d Z[b,i,d] * W[n,d]   (Z = concat(x,y))
//    one wave -> one 16x64 output strip (4 WMMA tiles share the A fragment).
// ---------------------------------------------------------------------------
__global__ void qkv_kernel(const float* __restrict__ x, const float* __restrict__ y,
                           const float* __restrict__ Wq, const float* __restrict__ Wk,
                           const float* __restrict__ Wv,
                           float* __restrict__ Q, float* __restrict__ K,
                           float* __restrict__ V) {
    const int lane = threadIdx.x & 31;
    const int gw = blockIdx.x * 8 + (threadIdx.x >> 5);
    const int tilesPerBatch = (MDIM / 16) * (DDIM / 64); // 32 * 4 = 128
    const int mat = gw / (NB_ALL * tilesPerBatch);
    const int rem = gw % (NB_ALL * tilesPerBatch);
    const int batch = rem / tilesPerBatch;
    const int t = rem % tilesPerBatch;
    const int rowTile = t / (DDIM / 64);
    const int cb = t % (DDIM / 64); // 64-wide column block

    const float* src = (batch < NB_X) ? (x + (long)batch * MD)
                                      : (y + (long)(batch - NB_X) * MD);
    const float* W = (mat == 0) ? Wq : (mat == 1) ? Wk : Wv;
    float* dst = ((mat == 0) ? Q : (mat == 1) ? K : V) + (long)batch * MD;

    const int m = lane & 15;   // A row / B col within tile
    const int kh = lane >> 4;  // K-half select
    const float* Arow = src + (long)(rowTile * 16 + m) * DDIM + 2 * kh;
    const float* Brow = W + (long)(cb * 64 + m) * DDIM + 2 * kh; // B[k][n]=W[n][k]

    v8f c0 = {}, c1 = {}, c2 = {}, c3 = {};
    for (int k0 = 0; k0 < DDIM; k0 += 4) {
        v2f a = *(const v2f*)(Arow + k0);
        v2f b0 = *(const v2f*)(Brow + k0);
        v2f b1 = *(const v2f*)(Brow + 16 * DDIM + k0);
        v2f b2 = *(const v2f*)(Brow + 32 * DDIM + k0);
        v2f b3 = *(const v2f*)(Brow + 48 * DDIM + k0);
        c0 = wmma_f32(a, b0, c0);
        c1 = wmma_f32(a, b1, c1);
        c2 = wmma_f32(a, b2, c2);
        c3 = wmma_f32(a, b3, c3);
    }
    const int col = cb * 64 + m;
    for (int r = 0; r < 8; ++r) {
        long row = rowTile * 16 + r + 8 * kh;
        float* dr = dst + row * DDIM + col;
        dr[0] = c0[r];
        dr[16] = c1[r];
        dr[32] = c2[r];
        dr[48] = c3[r];
    }
}

// ---------------------------------------------------------------------------
// 2) Masked scores: out[i,j] = mask ? Q[i,:].K[j,:] : NEG_FILL
//    mode 0: scores_x (64 pairs)  mode 1: scores_y (8)  mode 2: scores_xy (64)
//    one wave -> one 16x64 output strip.
// ---------------------------------------------------------------------------
__global__ void scores_kernel(const float* __restrict__ Q, const float* __restrict__ K,
                              const int* __restrict__ mask_x,
                              const int* __restrict__ mask_y,
                              float* __restrict__ out, int mode) {
    const int lane = threadIdx.x & 31;
    const int gw = blockIdx.x * 8 + (threadIdx.x >> 5);
    const int tilesPerPair = (MDIM / 16) * (MDIM / 64); // 32 * 8 = 256
    const int pair = gw / tilesPerPair;
    const int t = gw % tilesPerPair;
    const int rowTile = t / (MDIM / 64);
    const int cb = t % (MDIM / 64);

    int qb, kb;
    const int *mq, *mk;
    if (mode == 0) {
        qb = pair; kb = pair;
        mq = mask_x + pair * MDIM; mk = mq;
    } else if (mode == 1) {
        qb = NB_X + pair; kb = NB_X + pair;
        mq = mask_y + pair * MDIM; mk = mq;
    } else {
        const int n = pair >> 3, s = pair & 7;
        qb = s * 8 + n;            // Qx_e[n,s] = Qx[s*neg + n]
        kb = NB_X + s;             // Ky[s]
        mq = mask_x + qb * MDIM;
        mk = mask_y + s * MDIM;
    }

    const int m = lane & 15;
    const int kh = lane >> 4;
    const float* Arow = Q + (long)qb * MD + (long)(rowTile * 16 + m) * DDIM + 2 * kh;
    const float* Brow = K + (long)kb * MD + (long)(cb * 64 + m) * DDIM + 2 * kh;

    v8f c0 = {}, c1 = {}, c2 = {}, c3 = {};
    for (int k0 = 0; k0 < DDIM; k0 += 4) {
        v2f a = *(const v2f*)(Arow + k0);
        v2f b0 = *(const v2f*)(Brow + k0); // B[k][j] = K[j][k]
        v2f b1 = *(const v2f*)(Brow + 16 * DDIM + k0);
        v2f b2 = *(const v2f*)(Brow + 32 * DDIM + k0);
        v2f b3 = *(const v2f*)(Brow + 48 * DDIM + k0);
        c0 = wmma_f32(a, b0, c0);
        c1 = wmma_f32(a, b1, c1);
        c2 = wmma_f32(a, b2, c2);
        c3 = wmma_f32(a, b3, c3);
    }

    float* op = out + (long)pair * MDIM * MDIM;
    const int col = cb * 64 + m;
    const bool mk0 = (mk[col] != 0);
    const bool mk1 = (mk[col + 16] != 0);
    const bool mk2 = (mk[col + 32] != 0);
    const bool mk3 = (mk[col + 48] != 0);
    for (int r = 0; r < 8; ++r) {
        long row = rowTile * 16 + r + 8 * kh;
        const bool mr = (mq[row] != 0);
        float* orow = op + row * MDIM + col;
        orow[0] = (mr && mk0) ? c0[r] : NEG_FILLF;
        orow[16] = (mr && mk1) ? c1[r] : NEG_FILLF;
        orow[32] = (mr && mk2) ? c2[r] : NEG_FILLF;
        orow[48] = (mr && mk3) ? c3[r] : NEG_FILLF;
    }
}

// ---------------------------------------------------------------------------
// 3) mask_xy (as 0/1 float) and y_len (as float)
// ---------------------------------------------------------------------------
__global__ void mask_xy_kernel(const int* __restrict__ mask_x,
                               const int* __restrict__ mask_y,
                               float* __restrict__ out) {
    const long idx = (long)blockIdx.x * blockDim.x + threadIdx.x;
    const int j = (int)(idx & 511);
    const int i = (int)((idx >> 9) & 511);
    const int p = (int)(idx >> 18);
    const int s = p & 7, n = p >> 3;
    const bool v = (mask_x[(s * 8 + n) * MDIM + i] != 0) && (mask_y[s * MDIM + j] != 0);
    out[idx] = v ? 1.0f : 0.0f;
}

__global__ void ylen_kernel(const int* __restrict__ mask_y, float* __restrict__ out) {
    const int t = threadIdx.x; // 0..63 -> (n,s) with s = t&7
    const int s = t & 7;
    int cnt = 0;
    for (int j = 0; j < MDIM; ++j) cnt += (mask_y[s * MDIM + j] != 0);
    out[t] = (float)cnt;
}

// ---------------------------------------------------------------------------
// 4) Row stats (max, 1/sum(exp)) for scores_x + scores_y; wave per row.
// ---------------------------------------------------------------------------
__global__ void row_stats_kernel(const float* __restrict__ sx,
                                 const float* __restrict__ sy,
                                 float* __restrict__ smax, float* __restrict__ sisum) {
    const int lane = threadIdx.x & 31;
    const int gw = blockIdx.x * 8 + (threadIdx.x >> 5); // row id 0..36863
    const float* row = (gw < NB_X * MDIM) ? (sx + (long)gw * MDIM)
                                          : (sy + (long)(gw - NB_X * MDIM) * MDIM);
    float vals[16];
    float mx = -INFINITY;
    for (int t = 0; t < 16; ++t) {
        vals[t] = row[lane + t * 32];
        mx = fmaxf(mx, vals[t]);
    }
    for (int off = 16; off > 0; off >>= 1) mx = fmaxf(mx, __shfl_xor(mx, off, 32));
    float s = 0.0f;
    for (int t = 0; t < 16; ++t) s += expf(vals[t] - mx);
    for (int off = 16; off > 0; off >>= 1) s += __shfl_xor(s, off, 32);
    if (lane == 0) {
        smax[gw] = mx;
        sisum[gw] = 1.0f / s;
    }
}

// ---------------------------------------------------------------------------
// 5) In-place row softmax for probs_xy region; wave per row.
// ---------------------------------------------------------------------------
__global__ void softmax_rows_kernel(float* __restrict__ data) {
    const int lane = threadIdx.x & 31;
    const long gw = (long)blockIdx.x * 8 + (threadIdx.x >> 5);
    float* row = data + gw * MDIM;
    float vals[16];
    float mx = -INFINITY;
    for (int t = 0; t < 16; ++t) {
        vals[t] = row[lane + t * 32];
        mx = fmaxf(mx, vals[t]);
    }
    for (int off = 16; off > 0; off >>= 1) mx = fmaxf(mx, __shfl_xor(mx, off, 32));
    float s = 0.0f;
    for (int t = 0; t < 16; ++t) {
        vals[t] = expf(vals[t] - mx);
        s += vals[t];
    }
    for (int off = 16; off > 0; off >>= 1) s += __shfl_xor(s, off, 32);
    const float inv = 1.0f / s;
    for (int t = 0; t < 16; ++t) row[lane + t * 32] = vals[t] * inv;
}

// ---------------------------------------------------------------------------
// 6) ctx[i,d] = (1/sum_i) * sum_j exp(S[i,j]-m_i) * V[j,d]  via WMMA
//    one wave -> one 16x64 output strip; expf A-fragment amortized over 4 WMMAs.
// ---------------------------------------------------------------------------
__global__ void ctx_kernel(const float* __restrict__ S, const float* __restrict__ V,
                           const float* __restrict__ smax,
                           const float* __restrict__ sisum, float* __restrict__ out) {
    const int lane = threadIdx.x & 31;
    const int gw = blockIdx.x * 8 + (threadIdx.x >> 5);
    const int tilesPerBatch = (MDIM / 16) * (DDIM / 64); // 128
    const int batch = gw / tilesPerBatch;
    const int t = gw % tilesPerBatch;
    const int rowTile = t / (DDIM / 64);
    const int cb = t % (DDIM / 64);

    const int m = lane & 15;
    const int kh = lane >> 4;
    const float* Srow = S + (long)batch * MDIM * MDIM +
                        (long)(rowTile * 16 + m) * MDIM + 2 * kh;
    const float* Vb = V + (long)batch * MD + cb * 64 + m; // B[k][n] = V[k][n]
    const float mrow = smax[batch * MDIM + rowTile * 16 + m];

    v8f c0 = {}, c1 = {}, c2 = {}, c3 = {};
    for (int k0 = 0; k0 < MDIM; k0 += 4) {
        v2f a;
        a.x = expf(Srow[k0] - mrow);
        a.y = expf(Srow[k0 + 1] - mrow);
        const float* vk0 = Vb + (long)(k0 + 2 * kh) * DDIM;
        const float* vk1 = Vb + (long)(k0 + 2 * kh + 1) * DDIM;
        v2f b0, b1, b2, b3;
        b0.x = vk0[0];  b0.y = vk1[0];
        b1.x = vk0[16]; b1.y = vk1[16];
        b2.x = vk0[32]; b2.y = vk1[32];
        b3.x = vk0[48]; b3.y = vk1[48];
        c0 = wmma_f32(a, b0, c0);
        c1 = wmma_f32(a, b1, c1);
        c2 = wmma_f32(a, b2, c2);
        c3 = wmma_f32(a, b3, c3);
    }
    for (int r = 0; r < 8; ++r) {
        long row = rowTile * 16 + r + 8 * kh;
        const float isr = sisum[batch * MDIM + row];
        float* orow = out + (long)batch * MD + row * DDIM + cb * 64 + m;
        orow[0] = c0[r] * isr;
        orow[16] = c1[r] * isr;
        orow[32] = c2[r] * isr;
        orow[48] = c3[r] * isr;
    }
}

// ---------------------------------------------------------------------------
extern "C" void kernel_launch(void* const* d_in, const int* in_sizes, int n_in,
                              void* d_out, int out_size, void* d_ws, size_t ws_size,
                              hipStream_t stream) {
    const float* x = (const float*)d_in[0];
    const float* y = (const float*)d_in[1];
    const int* mask_x = (const int*)d_in[2];
    const int* mask_y = (const int*)d_in[3];
    const float* Wq = (const float*)d_in[4];
    const float* Wk = (const float*)d_in[5];
    const float* Wv = (const float*)d_in[6];
    float* out = (float*)d_out;
    float* ws = (float*)d_ws;

    float* Q = ws;
    float* K = ws + QKV_SZ;
    float* V = ws + 2 * QKV_SZ;
    float* smax = ws + 3 * QKV_SZ;        // 72*512 floats
    float* sisum = smax + NB_ALL * MDIM;  // 72*512 floats

    const dim3 blk(256);

    // 1) QKV projection: 3 * 72 * 128 strips = 27648 waves -> 3456 blocks
    qkv_kernel<<<3456, blk, 0, stream>>>(x, y, Wq, Wk, Wv, Q, K, V);

    // 2) scores: x (64*256 waves), y (8*256), xy (64*256 -> probs_xy region)
    scores_kernel<<<2048, blk, 0, stream>>>(Q, K, mask_x, mask_y, out + OFF_SX, 0);
    scores_kernel<<<256, blk, 0, stream>>>(Q, K, mask_x, mask_y, out + OFF_SY, 1);
    scores_kernel<<<2048, blk, 0, stream>>>(Q, K, mask_x, mask_y, out + OFF_PXY, 2);

    // 3) mask_xy (16.8M elems) and y_len
    mask_xy_kernel<<<65536, blk, 0, stream>>>(mask_x, mask_y, out + OFF_MXY);
    ylen_kernel<<<1, dim3(64), 0, stream>>>(mask_y, out + OFF_YLEN);

    // 4) row stats over scores_x + scores_y: 36864 rows -> 4608 blocks
    row_stats_kernel<<<4608, blk, 0, stream>>>(out + OFF_SX, out + OFF_SY, smax, sisum);

    // 5) softmax probs_xy in place: 32768 rows -> 4096 blocks
    softmax_rows_kernel<<<4096, blk, 0, stream>>>(out + OFF_PXY);

    // 6) ctx_x (64*128 strips) and ctx_y (8*128 strips)
    ctx_kernel<<<1024, blk, 0, stream>>>(out + OFF_SX, V, smax, sisum, out + OFF_CTX_X);
    ctx_kernel<<<128, blk, 0, stream>>>(out + OFF_SY, V + (long)NB_X * MD,
                                        smax + (long)NB_X * MDIM,
                                        sisum + (long)NB_X * MDIM, out + OFF_CTX_Y);
}